// NetRGCN_64252710748721
// MI455X (gfx1250) — compile-verified
//
#include <hip/hip_runtime.h>
#include <hip/hip_bf16.h>
#include <math.h>

// ---------------------------------------------------------------------------
// Problem constants (from reference): N=50000, C=256, R=5, E=800000,
// GLOBALS=25000, SENSES=20000. Output = concat(log_softmax_global[25000],
// log_softmax_sense[20000]).
// Only h[0] matters => edge scan for tgt==0, tiny per-relation mean, two big
// bandwidth-bound matvecs done with V_WMMA_F32_16X16X4_F32 tiles.
// ---------------------------------------------------------------------------

#define C_DIM 256
#define R_DIM 5

typedef float v2f __attribute__((ext_vector_type(2)));
typedef float v8f __attribute__((ext_vector_type(8)));

// ---- Kernel 0: zero the small accumulator region of the workspace ---------
__global__ void zero_ws_k(float* ws, int n) {
    int i = blockIdx.x * blockDim.x + threadIdx.x;
    if (i < n) ws[i] = 0.0f;
}

// ---- Kernel 1: scan edges, accumulate messages into target node 0 ---------
// agg[r*256+c] += x[src][c], cnt[r] += 1 for every edge with tgt==0.
__global__ void edge_scan_k(const int* __restrict__ edge_index,
                            const int* __restrict__ edge_type,
                            const float* __restrict__ x,
                            float* __restrict__ agg,
                            float* __restrict__ cnt,
                            int E) {
    int i = blockIdx.x * blockDim.x + threadIdx.x;
    if (i >= E) return;
    int tgt = edge_index[E + i];          // row 1 of (2,E)
    if (tgt == 0) {
        int src = edge_index[i];          // row 0 of (2,E)
        int r   = edge_type[i];
        atomicAdd(&cnt[r], 1.0f);
        const float* xs = x + (size_t)src * C_DIM;
        float* a = agg + r * C_DIM;
        for (int c = 0; c < C_DIM; ++c)
            atomicAdd(&a[c], xs[c]);
    }
}

// ---- Kernel 2: h0 = relu( sum_r mean_r @ W_rel[r] + x0 @ W_root + b ) -----
// One block, 1024 threads: thread = (d = tid&255, quarter = tid>>8).
__global__ void compute_h0_k(const float* __restrict__ x,
                             const float* __restrict__ W_rel,
                             const float* __restrict__ W_root,
                             const float* __restrict__ b_conv,
                             const float* __restrict__ agg,
                             const float* __restrict__ cnt,
                             float* __restrict__ h0) {
    __shared__ float mean[R_DIM * C_DIM];
    __shared__ float x0[C_DIM];
    __shared__ float part[4][C_DIM];
    int tid = threadIdx.x;

    for (int i = tid; i < R_DIM * C_DIM; i += 1024) {
        float c = cnt[i >> 8];
        mean[i] = agg[i] / fmaxf(c, 1.0f);
    }
    if (tid < C_DIM) x0[tid] = x[tid];
    __syncthreads();

    int d = tid & 255;
    int q = tid >> 8;
    float acc = 0.0f;
    // W_rel[(r*256+c)*256 + d], rc pairs striped over the 4 quarters
    for (int rc = q; rc < R_DIM * C_DIM; rc += 4)
        acc += mean[rc] * W_rel[(size_t)rc * C_DIM + d];
    for (int c = q; c < C_DIM; c += 4)
        acc += x0[c] * W_root[(size_t)c * C_DIM + d];
    part[q][d] = acc;
    __syncthreads();

    if (tid < C_DIM) {
        float v = part[0][tid] + part[1][tid] + part[2][tid] + part[3][tid]
                + b_conv[tid];
        h0[tid] = fmaxf(v, 0.0f);
    }
}

// ---- Kernel 3: logits = [Wg; Ws] @ h0 + [bg; bs] via f32 WMMA tiles -------
// One wave per 16-row tile. A = 16x4 W tile (ISA layout: lanes 0-15 K={0,1},
// lanes 16-31 K={2,3} => contiguous float2 per lane). B = h0 segment
// replicated across all 16 columns, so D's column 0 (lanes 0 and 16) holds
// the 16 dot products after the K-loop.
__global__ void matvec_wmma_k(const float* __restrict__ Wg,
                              const float* __restrict__ bg,
                              const float* __restrict__ Ws,
                              const float* __restrict__ bs,
                              const float* __restrict__ h0,
                              float* __restrict__ out,
                              int nG, int nS, int TG, int TS) {
    __shared__ float hs[C_DIM];
    int tid = threadIdx.x;
    hs[tid] = h0[tid];                 // blockDim.x == 256
    __syncthreads();

    int wave = tid >> 5;
    int lane = tid & 31;
    int gw = blockIdx.x * 8 + wave;    // 8 waves per block
    if (gw >= TG + TS) return;         // wave-uniform exit

    const float* W;  const float* bias;  float* o;  int nrows, rowbase;
    if (gw < TG) { W = Wg; bias = bg; o = out;      nrows = nG; rowbase = gw * 16; }
    else         { W = Ws; bias = bs; o = out + nG; nrows = nS; rowbase = (gw - TG) * 16; }

    int m     = lane & 15;             // A-matrix row within tile
    int khalf = (lane >> 4) << 1;      // 0 for lanes 0-15, 2 for lanes 16-31
    int row   = rowbase + m;
    if (row >= nrows) row = nrows - 1; // clamp load address; store is guarded
    const float* Arow = W + (size_t)row * C_DIM + khalf;

    v8f c = {0.f, 0.f, 0.f, 0.f, 0.f, 0.f, 0.f, 0.f};
    #pragma unroll 4
    for (int kb = 0; kb < C_DIM; kb += 4) {
        v2f a = *(const v2f*)(Arow + kb);            // global_load_b64, coalesced
        v2f b = *(const v2f*)(&hs[kb + khalf]);      // ds_load_b64
        c = __builtin_amdgcn_wmma_f32_16x16x4_f32(
                /*neg_a=*/false, a, /*neg_b=*/false, b,
                /*c_mod=*/(short)0, c, /*reuse_a=*/false, /*reuse_b=*/false);
    }

    // Column N==0 lives in lanes 0 (rows 0..7) and 16 (rows 8..15).
    if (m == 0) {
        int base = rowbase + ((lane >> 4) << 3);
        #pragma unroll
        for (int v = 0; v < 8; ++v) {
            int r2 = base + v;
            if (r2 < nrows) o[r2] = c[v] + bias[r2];
        }
    }
}

// ---- Kernel 4: in-place log_softmax per head (block 0: global, 1: sense) --
__global__ void log_softmax_k(float* __restrict__ out, int nG, int nS) {
    int head = blockIdx.x;
    float* p = out + (head ? nG : 0);
    int n    = head ? nS : nG;

    __shared__ float redA[32];
    __shared__ float redB[32];
    int tid = threadIdx.x;

    // pass 1: max
    float m = -INFINITY;
    for (int i = tid; i < n; i += 1024) m = fmaxf(m, p[i]);
    for (int o = 16; o > 0; o >>= 1) m = fmaxf(m, __shfl_xor(m, o, 32));
    if ((tid & 31) == 0) redA[tid >> 5] = m;
    __syncthreads();
    if (tid < 32) {
        float mm = redA[tid];
        for (int o = 16; o > 0; o >>= 1) mm = fmaxf(mm, __shfl_xor(mm, o, 32));
        if (tid == 0) redA[0] = mm;
    }
    __syncthreads();
    float gmax = redA[0];

    // pass 2: sum of exp
    float s = 0.0f;
    for (int i = tid; i < n; i += 1024) s += expf(p[i] - gmax);
    for (int o = 16; o > 0; o >>= 1) s += __shfl_xor(s, o, 32);
    if ((tid & 31) == 0) redB[tid >> 5] = s;
    __syncthreads();
    if (tid < 32) {
        float ss = redB[tid];
        for (int o = 16; o > 0; o >>= 1) ss += __shfl_xor(ss, o, 32);
        if (tid == 0) redB[0] = ss;
    }
    __syncthreads();
    float lse = logf(redB[0]);

    // pass 3: write result
    for (int i = tid; i < n; i += 1024) p[i] = p[i] - gmax - lse;
}

// ---------------------------------------------------------------------------
extern "C" void kernel_launch(void* const* d_in, const int* in_sizes, int n_in,
                              void* d_out, int out_size, void* d_ws, size_t ws_size,
                              hipStream_t stream) {
    const float* x      = (const float*)d_in[0];
    const int*   eidx   = (const int*)  d_in[1];
    const int*   etype  = (const int*)  d_in[2];
    const float* W_rel  = (const float*)d_in[3];
    const float* W_root = (const float*)d_in[4];
    const float* b_conv = (const float*)d_in[5];
    const float* W_g    = (const float*)d_in[6];
    const float* b_g    = (const float*)d_in[7];
    const float* W_s    = (const float*)d_in[8];
    const float* b_s    = (const float*)d_in[9];
    float* out = (float*)d_out;

    int E  = in_sizes[2];   // 800000
    int nG = in_sizes[7];   // 25000
    int nS = in_sizes[9];   // 20000

    // workspace layout (floats): agg[5*256] | cnt[5] pad to 8 | h0[256]
    float* ws  = (float*)d_ws;
    float* agg = ws;                      // 1280 floats
    float* cnt = ws + R_DIM * C_DIM;      // 5 floats
    float* h0  = ws + R_DIM * C_DIM + 8;  // 256 floats
    int zero_n = R_DIM * C_DIM + 8 + C_DIM;

    zero_ws_k<<<(zero_n + 255) / 256, 256, 0, stream>>>(ws, zero_n);
    edge_scan_k<<<(E + 255) / 256, 256, 0, stream>>>(eidx, etype, x, agg, cnt, E);
    compute_h0_k<<<1, 1024, 0, stream>>>(x, W_rel, W_root, b_conv, agg, cnt, h0);

    int TG = (nG + 15) / 16;
    int TS = (nS + 15) / 16;
    int waves  = TG + TS;
    int blocks = (waves + 7) / 8;         // 8 waves (256 threads) per block
    matvec_wmma_k<<<blocks, 256, 0, stream>>>(W_g, b_g, W_s, b_s, h0, out,
                                              nG, nS, TG, TS);

    log_softmax_k<<<2, 1024, 0, stream>>>(out, nG, nS);
}